// GCN_57174604644556
// MI455X (gfx1250) — compile-verified
//
#include <hip/hip_runtime.h>
#include <math.h>

// ---------------------------------------------------------------------------
// GCN forward for MI455X (gfx1250, wave32).
// GEMMs run on the fp32 matrix pipe via V_WMMA_F32_16X16X4_F32 (full fp32
// precision). W is staged through LDS in packed row-pairs so each B fragment
// is one ds_load_b64; K is a template parameter so the inner loop is fully
// branchless (tail handled by zero-filling B rows in LDS and clamping the A
// index). Edge aggregation is a wave-per-edge float4 gather +
// global_atomic_add_f32 scatter (HBM-bound, ~1.3 GB of traffic dominates).
// ---------------------------------------------------------------------------

typedef __attribute__((ext_vector_type(2))) float v2f;
typedef __attribute__((ext_vector_type(8))) float v8f;

#define DHID 128
#define TK 64
#define NEG_SLOPE 0.01f

// ---------------------------------------------------------------------------
// Generic zero fill (graph-capture-safe replacement for hipMemsetAsync).
// ---------------------------------------------------------------------------
__global__ void fill_zero(float* __restrict__ p, size_t n) {
    size_t i = blockIdx.x * (size_t)blockDim.x + threadIdx.x;
    if (i < n) p[i] = 0.0f;
}

// ---------------------------------------------------------------------------
// deg[dst] += 1 over all edges (self-loop +1 folded into rsqrt kernel).
// ---------------------------------------------------------------------------
__global__ void degree_kernel(const int* __restrict__ dst, float* __restrict__ deg, int E) {
    int e = blockIdx.x * blockDim.x + threadIdx.x;
    if (e < E) atomicAdd(deg + dst[e], 1.0f);
}

__global__ void rsqrt_kernel(const float* __restrict__ deg, float* __restrict__ dinv, int N) {
    int i = blockIdx.x * blockDim.x + threadIdx.x;
    if (i < N) dinv[i] = rsqrtf(deg[i] + 1.0f);  // +1 self loop
}

// ---------------------------------------------------------------------------
// WMMA fp32 GEMM: Out[N x 128] = act(A[N x K] @ W[K x 128] + bias)
// One wave32 computes a 16-row x 128-col slab (8 v8f accumulators); a block
// of 4 waves covers 64 rows and cooperatively stages W chunks into LDS.
//
// Fragment layouts (ISA 7.12.2, 32-bit, wave32):
//   A (16x4):  lane<16 -> M=lane,    K=k+{0,1};  lane>=16 -> M=lane-16, K=k+{2,3}
//   B (4x16):  N=lane&15, same K split across lane halves
//   C/D:       VGPR r, lanes 0-15 -> M=r, N=lane; lanes 16-31 -> M=r+8, N=lane-16
//
// LDS packing: W rows 2g, 2g+1 interleaved as Wp[g][col*2 + parity] so a B
// fragment (two K-rows, one column) is a single aligned v2f (ds_load_b64).
// ---------------------------------------------------------------------------
template <int K, bool BIAS, bool ACT>
__global__ __launch_bounds__(128) void gemm_wmma(
    const float* __restrict__ A, const float* __restrict__ W,
    const float* __restrict__ bias, float* __restrict__ Out, int Nrows)
{
    __shared__ float Wp[TK / 2][DHID * 2];   // 32 KB

    const int tid  = threadIdx.x;
    const int wave = tid >> 5;
    const int lane = tid & 31;
    const int row0 = (blockIdx.x * 4 + wave) * 16;
    const bool rowok = (row0 < Nrows);       // wave-uniform

    const int m     = lane & 15;             // M (A) / N (B) sub-index
    const int khalf = (lane >> 4) * 2;       // 0 or 2

    v8f acc[8];
#pragma unroll
    for (int t = 0; t < 8; ++t) acc[t] = (v8f){};

    const float* __restrict__ arow = A + (size_t)((rowok ? row0 : 0) + m) * K;

    for (int k0 = 0; k0 < K; k0 += TK) {
        // --- stage W[k0 .. k0+TK) into LDS (zero-fill rows >= K) -----------
#pragma unroll 8
        for (int idx = 0; idx < TK * DHID; idx += 128) {
            const int j   = idx + tid;
            const int krl = j >> 7;          // local row 0..TK-1
            const int col = j & (DHID - 1);
            const int kr  = k0 + krl;
            const float v = (kr < K) ? W[(size_t)kr * DHID + col] : 0.0f;
            Wp[krl >> 1][col * 2 + (krl & 1)] = v;
        }
        __syncthreads();

        // --- branchless WMMA sweep over this K chunk -----------------------
#pragma unroll 4
        for (int k = 0; k < TK; k += 4) {
            const int kk = k0 + k + khalf;
            // Clamp A index into bounds; B rows >= K are zero in LDS, so the
            // clamped (finite) A values contribute exactly 0.
            const int i0 = (kk     < K) ? kk     : (K - 1);
            const int i1 = (kk + 1 < K) ? kk + 1 : (K - 1);
            v2f a;
            a.x = arow[i0];
            a.y = arow[i1];
            const float* __restrict__ wp = &Wp[(k + khalf) >> 1][0];
#pragma unroll
            for (int t = 0; t < 8; ++t) {
                const v2f b = *(const v2f*)(wp + (t * 16 + m) * 2);
                acc[t] = __builtin_amdgcn_wmma_f32_16x16x4_f32(
                    false, a, false, b, (short)0, acc[t], false, false);
            }
        }
        __syncthreads();
    }

    if (!rowok) return;
    const int rbase = row0 + ((lane >> 4) ? 8 : 0);
#pragma unroll
    for (int t = 0; t < 8; ++t) {
        const int col = t * 16 + m;
        const float bv = BIAS ? bias[col] : 0.0f;
#pragma unroll
        for (int r = 0; r < 8; ++r) {
            float v = acc[t][r] + bv;
            if (ACT) v = (v >= 0.0f) ? v : NEG_SLOPE * v;
            Out[(size_t)(rbase + r) * DHID + col] = v;
        }
    }
}

// ---------------------------------------------------------------------------
// Edge scatter: agg[dst] += g[src] * (dinv[src]*dinv[dst]).
// One wave32 per edge; each lane moves one float4 (4 atomic f32 adds).
// ---------------------------------------------------------------------------
__global__ __launch_bounds__(256) void scatter_kernel(
    const float* __restrict__ g, const int* __restrict__ src,
    const int* __restrict__ dst, const float* __restrict__ dinv,
    float* __restrict__ agg, int E)
{
    const int t = blockIdx.x * blockDim.x + threadIdx.x;
    const int e = t >> 5;
    const int lane = t & 31;
    if (e >= E) return;
    const int s = src[e];
    const int d = dst[e];
    const float norm = dinv[s] * dinv[d];
    const float4 v = *(const float4*)(g + (size_t)s * DHID + lane * 4);
    float* ap = agg + (size_t)d * DHID + lane * 4;
    atomicAdd(ap + 0, v.x * norm);
    atomicAdd(ap + 1, v.y * norm);
    atomicAdd(ap + 2, v.z * norm);
    atomicAdd(ap + 3, v.w * norm);
}

// ---------------------------------------------------------------------------
// Finalize conv: out = agg + g * dinv^2 (self loop) + bias.  One thread per
// float4.
// ---------------------------------------------------------------------------
__global__ __launch_bounds__(256) void finalize_kernel(
    const float* __restrict__ agg, const float* __restrict__ g,
    const float* __restrict__ dinv, const float* __restrict__ bias,
    float* __restrict__ out, int Nrows)
{
    const size_t idx = blockIdx.x * (size_t)blockDim.x + threadIdx.x;
    const size_t total = (size_t)Nrows * (DHID / 4);
    if (idx >= total) return;
    const int i  = (int)(idx >> 5);        // node
    const int c4 = ((int)idx & 31) * 4;    // column group
    const float di = dinv[i];
    const float s = di * di;
    const float4 a  = *(const float4*)(agg + (size_t)i * DHID + c4);
    const float4 gg = *(const float4*)(g   + (size_t)i * DHID + c4);
    const float4 bb = *(const float4*)(bias + c4);
    float4 r;
    r.x = a.x + gg.x * s + bb.x;
    r.y = a.y + gg.y * s + bb.y;
    r.z = a.z + gg.z * s + bb.z;
    r.w = a.w + gg.w * s + bb.w;
    *(float4*)(out + (size_t)i * DHID + c4) = r;
}

// ---------------------------------------------------------------------------
// Output projection: out[N x 2] = H[N x 128] @ W2[128 x 2] + b2.
// Tiny GEMM -> one thread per row (512 B streamed per thread, W2 in L2).
// ---------------------------------------------------------------------------
__global__ __launch_bounds__(256) void out_layer_kernel(
    const float* __restrict__ H, const float* __restrict__ W2,
    const float* __restrict__ b2, float* __restrict__ out, int Nrows)
{
    const int i = blockIdx.x * blockDim.x + threadIdx.x;
    if (i >= Nrows) return;
    const float* h = H + (size_t)i * DHID;
    float a0 = 0.0f, a1 = 0.0f;
#pragma unroll 8
    for (int k = 0; k < DHID; ++k) {
        const float hv = h[k];
        a0 += hv * W2[k * 2 + 0];
        a1 += hv * W2[k * 2 + 1];
    }
    out[(size_t)i * 2 + 0] = a0 + b2[0];
    out[(size_t)i * 2 + 1] = a1 + b2[1];
}

// ---------------------------------------------------------------------------
// Launch pipeline (all on `stream`, graph-capture safe).
// ---------------------------------------------------------------------------
extern "C" void kernel_launch(void* const* d_in, const int* in_sizes, int n_in,
                              void* d_out, int out_size, void* d_ws, size_t ws_size,
                              hipStream_t stream)
{
    const float* x    = (const float*)d_in[0];
    const int*   ei   = (const int*)  d_in[1];   // [2, E] : src then dst
    // d_in[2] = edge_type (unused by reference)
    const float* W_in = (const float*)d_in[3];
    const float* b_in = (const float*)d_in[4];
    const float* W_g  = (const float*)d_in[5];
    const float* b_g  = (const float*)d_in[6];
    const float* W_o1 = (const float*)d_in[7];
    const float* b_o1 = (const float*)d_in[8];
    const float* W_o2 = (const float*)d_in[9];
    const float* b_o2 = (const float*)d_in[10];

    const int ND = 239;
    const int N  = in_sizes[0] / ND;   // 100000
    const int E  = in_sizes[1] / 2;    // 640000
    const int* src = ei;
    const int* dst = ei + E;

    // Workspace layout (floats): deg[N] | dinv[N] | bufA[N*128] | bufB[N*128] | bufC[N*128]
    float* ws   = (float*)d_ws;
    float* deg  = ws;
    float* dinv = ws + (size_t)N;
    float* bufA = ws + 2 * (size_t)N;              // h (layer input/output)
    float* bufB = bufA + (size_t)N * DHID;         // g = h @ W_g
    float* bufC = bufB + (size_t)N * DHID;         // agg (scatter target)

    const size_t nh = (size_t)N * DHID;
    const int gemm_blocks = (N / 16 + 3) / 4;
    const int scat_blocks = (int)(((size_t)E * 32 + 255) / 256);
    const int fin_blocks  = (int)((nh / 4 + 255) / 256);
    const int zero_blocks = (int)((nh + 255) / 256);

    // Degrees (shared by both convs)
    fill_zero<<<(N + 255) / 256, 256, 0, stream>>>(deg, (size_t)N);
    degree_kernel<<<(E + 255) / 256, 256, 0, stream>>>(dst, deg, E);
    rsqrt_kernel<<<(N + 255) / 256, 256, 0, stream>>>(deg, dinv, N);

    // h1 = LeakyReLU(x @ W_in + b_in)   [WMMA, K=239]
    gemm_wmma<239, true, true><<<gemm_blocks, 128, 0, stream>>>(x, W_in, b_in, bufA, N);

    // conv1: g = h1 @ W_g ; agg = scatter ; h2 = agg + g*dinv^2 + b_g
    gemm_wmma<128, false, false><<<gemm_blocks, 128, 0, stream>>>(bufA, W_g, nullptr, bufB, N);
    fill_zero<<<zero_blocks, 256, 0, stream>>>(bufC, nh);
    scatter_kernel<<<scat_blocks, 256, 0, stream>>>(bufB, src, dst, dinv, bufC, E);
    finalize_kernel<<<fin_blocks, 256, 0, stream>>>(bufC, bufB, dinv, b_g, bufA, N);

    // conv2 (same shared weights)
    gemm_wmma<128, false, false><<<gemm_blocks, 128, 0, stream>>>(bufA, W_g, nullptr, bufB, N);
    fill_zero<<<zero_blocks, 256, 0, stream>>>(bufC, nh);
    scatter_kernel<<<scat_blocks, 256, 0, stream>>>(bufB, src, dst, dinv, bufC, E);
    finalize_kernel<<<fin_blocks, 256, 0, stream>>>(bufC, bufB, dinv, b_g, bufA, N);

    // h4 = LeakyReLU(h3 @ W_o1 + b_o1)  [WMMA]
    gemm_wmma<128, true, true><<<gemm_blocks, 128, 0, stream>>>(bufA, W_o1, b_o1, bufB, N);

    // out = h4 @ W_o2 + b_o2
    out_layer_kernel<<<(N + 255) / 256, 256, 0, stream>>>(bufB, W_o2, b_o2, (float*)d_out, N);
}